// ProbAttention_49082886259025
// MI455X (gfx1250) — compile-verified
//
#include <hip/hip_runtime.h>
#include <math.h>

#define BH      32          // B*H
#define LSEQ    2048
#define DHEAD   64
#define NPAD    48          // n_top (40) padded to 3 tiles of 16
#define CHUNK   64          // keys per flash iteration
#define NCHUNK  (LSEQ / CHUNK)
#define LDSTR   68          // LDS row stride (floats) -> conflict-free

typedef __attribute__((ext_vector_type(2))) float v2f;
typedef __attribute__((ext_vector_type(8))) float v8f;

// ---------------------------------------------------------------------------
// Kernel 1: M[b,h,l] = max_s(Q·K_sample) - sum_s(Q·K_sample)/L
// one wave per query row; lane s handles samples s, s+32
// ---------------------------------------------------------------------------
__global__ __launch_bounds__(256) void sampled_m_kernel(
    const float* __restrict__ Q, const float* __restrict__ K,
    const int* __restrict__ idx, float* __restrict__ Mout, int SK) {
  int wave = threadIdx.x >> 5;
  int lane = threadIdx.x & 31;
  int gq = blockIdx.x * 8 + wave;        // [0, BH*LSEQ)
  int bh = gq >> 11;                     // / 2048
  int l  = gq & (LSEQ - 1);
  const float* qrow = Q + (size_t)gq * DHEAD;
  float mx = -3.0e38f;
  float sm = 0.0f;
  for (int s = lane; s < SK; s += 32) {
    int kidx = idx[l * SK + s];
    const float* krow = K + ((size_t)bh * LSEQ + kidx) * DHEAD;
    if (s + 32 < SK) {  // cover gather latency of the next sampled row
      int knext = idx[l * SK + s + 32];
      __builtin_prefetch(K + ((size_t)bh * LSEQ + knext) * DHEAD, 0, 3);
    }
    float acc = 0.0f;
#pragma unroll
    for (int d4 = 0; d4 < DHEAD / 4; ++d4) {
      float4 qv = ((const float4*)qrow)[d4];
      float4 kv = ((const float4*)krow)[d4];
      acc = fmaf(qv.x, kv.x, acc);
      acc = fmaf(qv.y, kv.y, acc);
      acc = fmaf(qv.z, kv.z, acc);
      acc = fmaf(qv.w, kv.w, acc);
    }
    mx = fmaxf(mx, acc);
    sm += acc;
  }
#pragma unroll
  for (int off = 16; off >= 1; off >>= 1) {
    mx = fmaxf(mx, __shfl_xor(mx, off, 32));
    sm += __shfl_xor(sm, off, 32);
  }
  if (lane == 0) Mout[gq] = mx - sm * (1.0f / (float)LSEQ);
}

// ---------------------------------------------------------------------------
// Kernel 2: per (b,h) iterative argmax top-k (ties -> lowest index, like jax)
// ---------------------------------------------------------------------------
__global__ __launch_bounds__(256) void topk_kernel(
    const float* __restrict__ Mvals, int* __restrict__ topk, int ntop) {
  __shared__ float vals[LSEQ];
  __shared__ float redv[256];
  __shared__ int   redi[256];
  int bh = blockIdx.x, tid = threadIdx.x;
  for (int i = tid; i < LSEQ; i += 256) vals[i] = Mvals[bh * LSEQ + i];
  __syncthreads();
  for (int it = 0; it < ntop; ++it) {
    float bv = -3.0e38f; int bi = 0x7fffffff;
    for (int i = tid; i < LSEQ; i += 256) {
      float v = vals[i];
      if (v > bv || (v == bv && i < bi)) { bv = v; bi = i; }
    }
    redv[tid] = bv; redi[tid] = bi;
    __syncthreads();
    for (int s = 128; s > 0; s >>= 1) {
      if (tid < s) {
        float v2 = redv[tid + s]; int i2 = redi[tid + s];
        if (v2 > redv[tid] || (v2 == redv[tid] && i2 < redi[tid])) {
          redv[tid] = v2; redi[tid] = i2;
        }
      }
      __syncthreads();
    }
    if (tid == 0) { topk[bh * ntop + it] = redi[0]; vals[redi[0]] = -3.0e38f; }
    __syncthreads();
  }
}

// ---------------------------------------------------------------------------
// Kernel 3: mean of V over L per (b,h,d)
// ---------------------------------------------------------------------------
__global__ __launch_bounds__(256) void vmean_kernel(
    const float* __restrict__ V, float* __restrict__ mean) {
  __shared__ float red[256];
  int bh = blockIdx.x, tid = threadIdx.x;
  int d = tid & 63, g = tid >> 6;
  float s = 0.0f;
  for (int l = g; l < LSEQ; l += 4)
    s += V[((size_t)bh * LSEQ + l) * DHEAD + d];
  red[tid] = s;
  __syncthreads();
  if (g == 0)
    mean[bh * DHEAD + d] =
        (red[d] + red[64 + d] + red[128 + d] + red[192 + d]) * (1.0f / LSEQ);
}

// ---------------------------------------------------------------------------
// Kernel 4: broadcast mean into the full output (float4, fully coalesced)
// ---------------------------------------------------------------------------
__global__ __launch_bounds__(256) void fill_kernel(
    const float* __restrict__ mean, float4* __restrict__ out) {
  int i = blockIdx.x * 256 + threadIdx.x;   // [0, BH*LSEQ*16)
  int d4 = i & 15;                          // float4 index within D
  int bh = i >> 15;                         // / (LSEQ*16)
  out[i] = *(const float4*)&mean[bh * DHEAD + d4 * 4];
}

// ---------------------------------------------------------------------------
// Kernel 5: WMMA flash attention over the 40 selected query rows per (b,h).
// 12 waves: wave = (row_tile rt in 0..2) x (col_tile ct in 0..3).
// K/V chunks are double-buffered and streamed with async global->LDS loads
// (ASYNCcnt), overlapping the next chunk's fetch with WMMA + softmax.
// ---------------------------------------------------------------------------
__global__ __launch_bounds__(384) void prob_attn_kernel(
    const float* __restrict__ Q, const float* __restrict__ K,
    const float* __restrict__ V, const int* __restrict__ topk,
    float* __restrict__ out, int ntop) {
  __shared__ float Qs[NPAD * LDSTR];
  __shared__ float Ks[2][CHUNK * LDSTR];
  __shared__ float Vs[2][CHUNK * LDSTR];
  __shared__ float Ps[NPAD * LDSTR];
  __shared__ float mrow[NPAD], lrow[NPAD], crow[NPAD];

  int bh = blockIdx.x, tid = threadIdx.x;
  int wave = tid >> 5, lane = tid & 31;
  int rt = wave >> 2, ct = wave & 3;
  int m15 = lane & 15, kh = lane >> 4;

  // Issue async global->LDS B128 copies for one K/V chunk into buffer `buf`.
  auto issue_chunk = [&](int chunkIdx, int buf) {
    int nb = chunkIdx * CHUNK;
    for (int i = tid; i < CHUNK * (DHEAD / 4); i += 384) {
      int row = i >> 4, c4 = (i & 15) * 4;
      const float* kp = K + ((size_t)bh * LSEQ + nb + row) * DHEAD + c4;
      const float* vp = V + ((size_t)bh * LSEQ + nb + row) * DHEAD + c4;
      // Low 32 bits of a generic LDS pointer are the LDS byte offset.
      unsigned lk = (unsigned)(uintptr_t)&Ks[buf][row * LDSTR + c4];
      unsigned lv = (unsigned)(uintptr_t)&Vs[buf][row * LDSTR + c4];
      asm volatile("global_load_async_to_lds_b128 %0, %1, off"
                   :: "v"(lk), "v"(kp) : "memory");
      asm volatile("global_load_async_to_lds_b128 %0, %1, off"
                   :: "v"(lv), "v"(vp) : "memory");
    }
  };

  { // stage Q_reduce rows (zero-padded to 48)
    int r  = tid >> 3;           // 0..47
    int c0 = (tid & 7) * 8;      // 0..56
    float4 a = {0, 0, 0, 0}, b = {0, 0, 0, 0};
    if (r < ntop) {
      int qidx = topk[bh * ntop + r];
      const float4* src =
          (const float4*)(Q + ((size_t)bh * LSEQ + qidx) * DHEAD + c0);
      a = src[0]; b = src[1];
    }
    float* dst = &Qs[r * LDSTR + c0];
    ((float4*)dst)[0] = a; ((float4*)dst)[1] = b;
    if (tid < NPAD) { mrow[tid] = -3.0e38f; lrow[tid] = 0.0f; }
  }

  issue_chunk(0, 0);                                   // prefetch chunk 0
  asm volatile("s_wait_asynccnt 0x0" ::: "memory");
  __syncthreads();                                     // Q + chunk 0 visible

  v8f o_acc = {0.f, 0.f, 0.f, 0.f, 0.f, 0.f, 0.f, 0.f};
  const float scale = 0.125f;  // 1/sqrt(64)

  for (int chunk = 0; chunk < NCHUNK; ++chunk) {
    int cur = chunk & 1;
    if (chunk + 1 < NCHUNK)                            // stream next chunk
      issue_chunk(chunk + 1, (chunk + 1) & 1);

    // S tile: Q(16xD) x K^T(Dx16), D=64 -> 16 f32 WMMAs
    v8f s_acc = {0.f, 0.f, 0.f, 0.f, 0.f, 0.f, 0.f, 0.f};
#pragma unroll
    for (int step = 0; step < DHEAD / 4; ++step) {
      int off = step * 4 + 2 * kh;
      v2f a  = *(const v2f*)&Qs[(rt * 16 + m15) * LDSTR + off];
      v2f bm = *(const v2f*)&Ks[cur][(ct * 16 + m15) * LDSTR + off];
      s_acc = __builtin_amdgcn_wmma_f32_16x16x4_f32(
          false, a, false, bm, (short)0, s_acc, false, false);
    }
#pragma unroll
    for (int v = 0; v < 8; ++v)
      Ps[(rt * 16 + v + 8 * kh) * LDSTR + ct * 16 + m15] = s_acc[v] * scale;
    __syncthreads();

    // online softmax row update (48 rows, one thread each)
    if (tid < NPAD) {
      float cm = -3.0e38f;
      for (int c = 0; c < CHUNK; ++c) cm = fmaxf(cm, Ps[tid * LDSTR + c]);
      float mold = mrow[tid];
      float mnew = fmaxf(mold, cm);
      float corr = __expf(mold - mnew);
      float s = 0.0f;
      for (int c = 0; c < CHUNK; ++c) {
        float p = __expf(Ps[tid * LDSTR + c] - mnew);
        Ps[tid * LDSTR + c] = p;
        s += p;
      }
      lrow[tid] = lrow[tid] * corr + s;
      mrow[tid] = mnew;
      crow[tid] = corr;
    }
    __syncthreads();

    // rescale accumulator, then O += P(16x64) x V(64x16): 16 f32 WMMAs
#pragma unroll
    for (int v = 0; v < 8; ++v) o_acc[v] *= crow[rt * 16 + v + 8 * kh];
#pragma unroll
    for (int step = 0; step < CHUNK / 4; ++step) {
      int noff = step * 4 + 2 * kh;
      v2f a = *(const v2f*)&Ps[(rt * 16 + m15) * LDSTR + noff];
      v2f bm;
      bm.x = Vs[cur][noff * LDSTR + ct * 16 + m15];
      bm.y = Vs[cur][(noff + 1) * LDSTR + ct * 16 + m15];
      o_acc = __builtin_amdgcn_wmma_f32_16x16x4_f32(
          false, a, false, bm, (short)0, o_acc, false, false);
    }

    asm volatile("s_wait_asynccnt 0x0" ::: "memory");  // next chunk landed
    __syncthreads();
  }

  // scatter normalized rows into the output
#pragma unroll
  for (int v = 0; v < 8; ++v) {
    int r = rt * 16 + v + 8 * kh;
    if (r < ntop) {
      int qidx = topk[bh * ntop + r];
      out[((size_t)bh * LSEQ + qidx) * DHEAD + ct * 16 + m15] =
          o_acc[v] / lrow[r];
    }
  }
}

// ---------------------------------------------------------------------------
extern "C" void kernel_launch(void* const* d_in, const int* in_sizes, int n_in,
                              void* d_out, int out_size, void* d_ws,
                              size_t ws_size, hipStream_t stream) {
  const float* Q  = (const float*)d_in[0];
  const float* K  = (const float*)d_in[1];
  const float* V  = (const float*)d_in[2];
  const int* idx  = (const int*)d_in[3];
  float* out = (float*)d_out;

  int SK   = in_sizes[3] / LSEQ;   // sample_k (== 40)
  int ntop = SK;                   // n_top uses the same formula
  if (ntop > NPAD) ntop = NPAD;

  char* ws = (char*)d_ws;
  float* Mbuf = (float*)ws;                                   // BH*LSEQ floats
  size_t off = (size_t)BH * LSEQ * sizeof(float);
  int* topk = (int*)(ws + off);                               // BH*ntop ints
  off += ((size_t)BH * ntop * sizeof(int) + 255) & ~(size_t)255;
  float* mean = (float*)(ws + off);                           // BH*DHEAD floats

  sampled_m_kernel<<<(BH * LSEQ) / 8, 256, 0, stream>>>(Q, K, idx, Mbuf, SK);
  topk_kernel<<<BH, 256, 0, stream>>>(Mbuf, topk, ntop);
  vmean_kernel<<<BH, 256, 0, stream>>>(V, mean);
  fill_kernel<<<(BH * LSEQ * DHEAD / 4) / 256, 256, 0, stream>>>(
      mean, (float4*)out);
  prob_attn_kernel<<<BH, 384, 0, stream>>>(Q, K, V, topk, out, ntop);
}